// AttentionConv3D_82282983457190
// MI455X (gfx1250) — compile-verified
//
#include <hip/hip_runtime.h>
#include <hip/hip_bf16.h>

typedef __attribute__((ext_vector_type(2))) float v2f;
typedef __attribute__((ext_vector_type(8))) float v8f;

#define CIN   64
#define COUT  64
#define DDIM  16
#define HDIM  64
#define WDIM  64
#define NPOS  (DDIM * HDIM * WDIM)   // 65536

// ---------------------------------------------------------------------------
// Kernel 1: q/k/v = W{q,k,v} @ x   via V_WMMA_F32_16X16X4_F32 (exact f32).
// Block = 256 threads = 8 waves; block covers 32 positions, all 64 out chans.
// Wave w: M-tile = w&3 (16 output channels), N-tile = w>>2 (16 positions).
// ---------------------------------------------------------------------------
__global__ __launch_bounds__(256) void qkv_wmma_kernel(
    const float* __restrict__ x,    // [CIN, NPOS]
    const float* __restrict__ wq,   // [COUT, CIN] row-major
    const float* __restrict__ wk,
    const float* __restrict__ wv,
    float* __restrict__ q,          // [COUT, NPOS]
    float* __restrict__ k,
    float* __restrict__ v)
{
    __shared__ float xs[CIN * 32];  // x tile: [c][32 positions], 8 KB

    const int tid   = threadIdx.x;
    const int nbase = blockIdx.x * 32;

    // Cooperative load of the x tile into LDS (128B contiguous per row chunk).
    #pragma unroll
    for (int i = 0; i < 8; ++i) {
        int idx = tid + 256 * i;          // 0..2047
        int c   = idx >> 5;
        int j   = idx & 31;
        xs[idx] = x[(size_t)c * NPOS + (size_t)(nbase + j)];
    }
    __syncthreads();

    const int wave  = tid >> 5;
    const int lane  = tid & 31;
    const int mtile = wave & 3;
    const int ntile = wave >> 2;          // 0 or 1
    const int m0    = mtile << 4;
    const int l15   = lane & 15;
    const int koff  = (lane < 16) ? 0 : 2;  // K-half held by this lane (A and B)
    const int mrow  = m0 + l15;             // A: this lane's M row
    const int nl    = (ntile << 4) + l15;   // B: this lane's column in the tile

    v8f accq = {}; v8f acck = {}; v8f accv = {};

    #pragma unroll
    for (int kk = 0; kk < 16; ++kk) {
        const int c0 = (kk << 2) + koff;    // base input channel for this lane

        // B operand (4x16 f32): V0 = x[c0, n], V1 = x[c0+1, n]
        v2f b;
        b.x = xs[(c0    ) * 32 + nl];
        b.y = xs[(c0 + 1) * 32 + nl];

        // A operand (16x4 f32): V0 = w[m, c0], V1 = w[m, c0+1]
        v2f a;
        a.x = wq[mrow * CIN + c0];
        a.y = wq[mrow * CIN + c0 + 1];
        accq = __builtin_amdgcn_wmma_f32_16x16x4_f32(
                   false, a, false, b, (short)0, accq, false, false);

        a.x = wk[mrow * CIN + c0];
        a.y = wk[mrow * CIN + c0 + 1];
        acck = __builtin_amdgcn_wmma_f32_16x16x4_f32(
                   false, a, false, b, (short)0, acck, false, false);

        a.x = wv[mrow * CIN + c0];
        a.y = wv[mrow * CIN + c0 + 1];
        accv = __builtin_amdgcn_wmma_f32_16x16x4_f32(
                   false, a, false, b, (short)0, accv, false, false);
    }

    // D layout (16x16 f32): VGPR r holds M=r (lanes 0-15) / M=r+8 (lanes 16-31),
    // N = lane % 16.
    const int    mhalf = (lane < 16) ? 0 : 8;
    const size_t ncol  = (size_t)(nbase + nl);
    #pragma unroll
    for (int r = 0; r < 8; ++r) {
        size_t off = (size_t)(m0 + r + mhalf) * NPOS + ncol;
        q[off] = accq[r];
        k[off] = acck[r];
        v[off] = accv[r];
    }
}

// ---------------------------------------------------------------------------
// Kernel 2: per-channel 3x3x3 windowed softmax-attention.
// One thread per (c,d,h,w); lanes contiguous in w -> coalesced tap loads.
// Zero-padded k/v (bias still applied to padded taps, matching reference).
// ---------------------------------------------------------------------------
__global__ __launch_bounds__(256) void attn_kernel(
    const float* __restrict__ q,
    const float* __restrict__ k,
    const float* __restrict__ v,
    const float* __restrict__ rel_d,   // [21][3]
    const float* __restrict__ rel_h,   // [21][3]
    const float* __restrict__ rel_w,   // [22][3]
    float* __restrict__ out)
{
    const int t = blockIdx.x * 256 + threadIdx.x;   // == c*NPOS + d*4096 + h*64 + w
    const int w = t & 63;
    const int h = (t >> 6) & 63;
    const int d = (t >> 12) & 15;
    const int c = t >> 16;

    // Per-channel rel bias: axis 0 -> depends on td, 1 -> th, 2 -> tw.
    float b0, b1, b2;
    int axis;
    if (c < 21)      { axis = 0; const float* r = rel_d + c * 3;        b0 = r[0]; b1 = r[1]; b2 = r[2]; }
    else if (c < 42) { axis = 1; const float* r = rel_h + (c - 21) * 3; b0 = r[0]; b1 = r[1]; b2 = r[2]; }
    else             { axis = 2; const float* r = rel_w + (c - 42) * 3; b0 = r[0]; b1 = r[1]; b2 = r[2]; }

    const size_t cbase = (size_t)c * NPOS;
    const float* kc = k + cbase;
    const float* vc = v + cbase;
    const int    p  = ((d * HDIM) + h) * WDIM + w;
    const float  qv = q[cbase + p];

    float logit[27], vtap[27];
    float mx = -3.402823466e+38f;
    int   ti = 0;

    #pragma unroll
    for (int td = 0; td < 3; ++td) {
        #pragma unroll
        for (int th = 0; th < 3; ++th) {
            #pragma unroll
            for (int tw = 0; tw < 3; ++tw) {
                const int dd = d + td - 1;
                const int hh = h + th - 1;
                const int ww = w + tw - 1;
                const bool inb = ((unsigned)dd < (unsigned)DDIM) &
                                 ((unsigned)hh < (unsigned)HDIM) &
                                 ((unsigned)ww < (unsigned)WDIM);
                const int off  = ((dd * HDIM) + hh) * WDIM + ww;
                const int offc = inb ? off : 0;               // clamp: no OOB reads
                const float kraw = kc[offc];
                const float vraw = vc[offc];
                const float kv = inb ? kraw : 0.0f;           // zero-pad
                const float vv = inb ? vraw : 0.0f;
                const int   bi = (axis == 0) ? td : ((axis == 1) ? th : tw);
                const float bias = (bi == 0) ? b0 : ((bi == 1) ? b1 : b2);
                const float lg = qv * (kv + bias);
                logit[ti] = lg;
                vtap[ti]  = vv;
                mx = fmaxf(mx, lg);
                ++ti;
            }
        }
    }

    float ssum = 0.0f, osum = 0.0f;
    #pragma unroll
    for (int i = 0; i < 27; ++i) {
        const float e = __expf(logit[i] - mx);
        ssum += e;
        osum += e * vtap[i];
    }
    out[t] = osum / ssum;
}

// ---------------------------------------------------------------------------
extern "C" void kernel_launch(void* const* d_in, const int* in_sizes, int n_in,
                              void* d_out, int out_size, void* d_ws, size_t ws_size,
                              hipStream_t stream)
{
    const float* x     = (const float*)d_in[0];
    const float* wq    = (const float*)d_in[1];
    const float* wk    = (const float*)d_in[2];
    const float* wv    = (const float*)d_in[3];
    const float* rel_d = (const float*)d_in[4];
    const float* rel_h = (const float*)d_in[5];
    const float* rel_w = (const float*)d_in[6];

    float* qb = (float*)d_ws;                    // [64, 65536] f32
    float* kb = qb + (size_t)COUT * NPOS;
    float* vb = kb + (size_t)COUT * NPOS;

    qkv_wmma_kernel<<<NPOS / 32, 256, 0, stream>>>(x, wq, wk, wv, qb, kb, vb);
    attn_kernel<<<(COUT * NPOS) / 256, 256, 0, stream>>>(
        qb, kb, vb, rel_d, rel_h, rel_w, (float*)d_out);
}